// Ss_rnn_77163382440496
// MI455X (gfx1250) — compile-verified
//
#include <hip/hip_runtime.h>

// MI455X / gfx1250, wave32. All matrix math in f32 via V_WMMA_F32_16X16X4_F32
// (16x16 tiles, K stepped by 4) to keep full precision across the 262144-step
// linear recurrence, which we parallelize by two levels of chunking:
//   level 1: 16 steps/chunk   -> dense GEMMs over u_proj
//   level 2: 16 chunks/super  -> dense GEMM with block-triangular operator
//   residual: 1024 dependent 64x64 matvecs (tiny sequential kernel)

typedef __attribute__((ext_vector_type(2))) float v2f;
typedef __attribute__((ext_vector_type(8))) float v8f;

#define T_STEPS 262144
#define NCHUNK  16384   // T/16
#define NSUPER  1024    // NCHUNK/16

// ---------------------------------------------------------------------------
// 16x(16*NT)-strip WMMA f32 GEMM: D = A(MxK) @ B(KxN) [+ Cadd] [+ bias[n]]
// One wave per 16x(16*NT) output strip: the A fragment is loaded once per
// K-step and reused across NT back-to-back v_wmma (distinct accumulators, no
// D->A/B hazard). M,N multiples of 16*NT in the launched dims, K of 4.
// VGPR layouts per CDNA5 ISA 7.12.2:
//   A 16x4 f32 : lane m = l&15; VGPR0 = K {0|2}, VGPR1 = K {1|3} by lane half
//   B 4x16 f32 : lane n = l&15; rows striped across lanes, K split by half
//   C/D 16x16  : VGPR r -> row r + 8*(l>=16), col l&15
// ---------------------------------------------------------------------------
template <int NT>
__global__ __launch_bounds__(32)
void rnnscan_gemm_wmma(const float* __restrict__ A, const float* __restrict__ B,
                       const float* __restrict__ Cadd, const float* __restrict__ biasN,
                       float* __restrict__ D, int M, int N, int K)
{
  const int l   = threadIdx.x;   // 0..31
  const int n16 = l & 15;
  const int hi  = l >> 4;        // 0 or 1
  const int kb  = hi * 2;        // K offset handled by this lane half
  const int mt  = blockIdx.x;
  const int nt0 = blockIdx.y * NT;

  const float* __restrict__ Arow = A + (size_t)(mt * 16 + n16) * K; // lane's A row
  const float* __restrict__ Bcol = B + (size_t)nt0 * 16 + n16;       // lane's B col base

  v8f acc[NT];
  #pragma unroll
  for (int t = 0; t < NT; ++t) acc[t] = (v8f){0.f,0.f,0.f,0.f,0.f,0.f,0.f,0.f};

  for (int k0 = 0; k0 < K; k0 += 4) {
    const v2f a = *(const v2f*)(Arow + k0 + kb);   // A[m][k0+kb..+1] (8B aligned)
    const float* __restrict__ Br0 = Bcol + (size_t)(k0 + kb) * N;
    const float* __restrict__ Br1 = Bcol + (size_t)(k0 + kb + 1) * N;
    #pragma unroll
    for (int t = 0; t < NT; ++t) {
      v2f b;
      b.x = Br0[t * 16];                            // B[k0+kb  ][n + 16t]
      b.y = Br1[t * 16];                            // B[k0+kb+1][n + 16t]
      // 8 args: (neg_a, A, neg_b, B, c_mod, C, reuse_a, reuse_b)
      acc[t] = __builtin_amdgcn_wmma_f32_16x16x4_f32(false, a, false, b,
                                                     (short)0, acc[t], false, false);
    }
  }

  #pragma unroll
  for (int t = 0; t < NT; ++t) {
    const int   col  = (nt0 + t) * 16 + n16;
    const float bias = biasN ? biasN[col] : 0.f;
    #pragma unroll
    for (int r = 0; r < 8; ++r) {
      const int    row = mt * 16 + r + hi * 8;
      const size_t idx = (size_t)row * N + col;
      float v = acc[t][r] + bias;
      if (Cadd) v += Cadd[idx];
      D[idx] = v;
    }
  }
}

// ---------------------------------------------------------------------------
// Setup: build A = Ws, WuT, powers A^k and (A^16)^k, q_k = w_y^T A^k, and the
// packed operator matrices used by the GEMM phases. One WG, runs in ~us.
// ---------------------------------------------------------------------------
__device__ inline void rnnscan_mm64(const float* __restrict__ A,
                                    const float* __restrict__ B,
                                    float* __restrict__ C)
{
  for (int idx = threadIdx.x; idx < 64 * 64; idx += 256) {
    const int i = idx >> 6, j = idx & 63;
    float s = 0.f;
    #pragma unroll 8
    for (int k = 0; k < 64; ++k) s = fmaf(A[i * 64 + k], B[k * 64 + j], s);
    C[idx] = s;
  }
}

__global__ __launch_bounds__(256)
void rnnscan_setup(const float* __restrict__ W_u2x, const float* __restrict__ b_u2x,
                   const float* __restrict__ W_x2y, const float* __restrict__ b_x2y,
                   float* Amat, float* WuT, float* Apow, float* A16pow, float* qv,
                   float* Gt, float* Lt, float* Mt, float* Rt, float* Q1t, float* by16)
{
  const int tid = threadIdx.x;

  // A = Ws = W_u2x[:, 16:80];  WuT[k][n] = W_u2x[n][k] (input projection, KxN)
  for (int idx = tid; idx < 64 * 64; idx += 256) {
    const int i = idx >> 6, j = idx & 63;
    Amat[idx] = W_u2x[i * 80 + 16 + j];
  }
  for (int idx = tid; idx < 16 * 64; idx += 256) {
    const int k = idx >> 6, n = idx & 63;
    WuT[idx] = W_u2x[n * 80 + k];
  }
  // Apow[0] = I
  for (int idx = tid; idx < 64 * 64; idx += 256) {
    const int i = idx >> 6, j = idx & 63;
    Apow[idx] = (i == j) ? 1.f : 0.f;
  }
  __syncthreads();
  for (int idx = tid; idx < 64 * 64; idx += 256) Apow[4096 + idx] = Amat[idx];
  __syncthreads();
  for (int k = 2; k <= 16; ++k) {              // A^k = A^(k-1) @ A
    rnnscan_mm64(Apow + (size_t)(k - 1) * 4096, Amat, Apow + (size_t)k * 4096);
    __syncthreads();
  }
  // A16pow[0] = I, A16pow[1] = A^16
  for (int idx = tid; idx < 64 * 64; idx += 256) {
    const int i = idx >> 6, j = idx & 63;
    A16pow[idx]        = (i == j) ? 1.f : 0.f;
    A16pow[4096 + idx] = Apow[16 * 4096 + idx];
  }
  __syncthreads();
  for (int k = 2; k <= 16; ++k) {              // (A^16)^k
    rnnscan_mm64(A16pow + (size_t)(k - 1) * 4096, A16pow + 4096,
                 A16pow + (size_t)k * 4096);
    __syncthreads();
  }
  // q_k[b] = sum_a w_y[a] * A^k[a][b],  k = 0..16
  for (int idx = tid; idx < 17 * 64; idx += 256) {
    const int k = idx >> 6, b = idx & 63;
    float s = 0.f;
    for (int a = 0; a < 64; ++a) s = fmaf(W_x2y[a], Apow[k * 4096 + a * 64 + b], s);
    qv[idx] = s;
  }
  __syncthreads();
  // Gt[(j*64+b)][a] = A^(15-j)[a][b]          (chunk carry operator, 1024x64)
  for (int idx = tid; idx < 1024 * 64; idx += 256) {
    const int row = idx >> 6, a = idx & 63;
    const int j = row >> 6, b = row & 63;
    Gt[idx] = Apow[(15 - j) * 4096 + a * 64 + b];
  }
  // Lt[(j*64+b)][i] = (j<=i) ? q_{i-j}[b] : 0 (local output operator, 1024x16)
  for (int idx = tid; idx < 1024 * 16; idx += 256) {
    const int row = idx >> 4, i = idx & 15;
    const int j = row >> 6, b = row & 63;
    Lt[idx] = (j <= i) ? qv[(i - j) * 64 + b] : 0.f;
  }
  // Mt[(j*64+b)][(i*64+a)] = (j<=i) ? A16^(i-j)[a][b] : 0  (block-tri, 1024x1024)
  for (int idx = tid; idx < 1024 * 1024; idx += 256) {
    const int row = idx >> 10, col = idx & 1023;
    const int j = row >> 6, b = row & 63;
    const int i = col >> 6, a = col & 63;
    Mt[idx] = (j <= i) ? A16pow[(i - j) * 4096 + a * 64 + b] : 0.f;
  }
  // Rt[b][(i*64+a)] = A16^(i+1)[a][b]         (carry broadcast, 64x1024)
  for (int idx = tid; idx < 64 * 1024; idx += 256) {
    const int b = idx >> 10, col = idx & 1023;
    const int i = col >> 6, a = col & 63;
    Rt[idx] = A16pow[(i + 1) * 4096 + a * 64 + b];
  }
  // Q1t[b][i] = q_{i+1}[b]                     (output projection, 64x16)
  for (int idx = tid; idx < 64 * 16; idx += 256) {
    const int b = idx >> 4, i = idx & 15;
    Q1t[idx] = qv[(i + 1) * 64 + b];
  }
  if (tid < 16) by16[tid] = b_x2y[0];
}

// ---------------------------------------------------------------------------
// Residual sequential scan over 1024 super-chunks: S = A256 @ S + d_sc,
// d_sc = SlocFlat[sc][960..1023]. Writes S_{sc-1} (state entering sc).
// ---------------------------------------------------------------------------
__global__ __launch_bounds__(64)
void rnnscan_scan(const float* __restrict__ A256, const float* __restrict__ SlocFlat,
                  const float* __restrict__ state_init, float* __restrict__ Sbound)
{
  __shared__ float s[64];
  const int t = threadIdx.x;
  s[t] = state_init[t];
  __syncthreads();
  const float* __restrict__ Ar = A256 + t * 64;
  for (int sc = 0; sc < NSUPER; ++sc) {
    Sbound[sc * 64 + t] = s[t];                       // S_{sc-1}
    float acc = SlocFlat[(size_t)sc * 1024 + 960 + t];
    #pragma unroll 8
    for (int b = 0; b < 64; ++b) acc = fmaf(Ar[b], s[b], acc);
    __syncthreads();
    s[t] = acc;
    __syncthreads();
  }
}

// StatesPrev[c] = state entering chunk c = States[c-1] (init state for c=0)
__global__ void rnnscan_statesprev(const float* __restrict__ StateFlat,
                                   const float* __restrict__ state_init,
                                   float* __restrict__ SP)
{
  const int idx = blockIdx.x * 256 + threadIdx.x;
  if (idx >= NCHUNK * 64) return;
  const int c = idx >> 6, a = idx & 63;
  SP[idx] = (c == 0) ? state_init[a]
                     : StateFlat[(size_t)((c - 1) >> 4) * 1024 + ((c - 1) & 15) * 64 + a];
}

// ---------------------------------------------------------------------------
extern "C" void kernel_launch(void* const* d_in, const int* in_sizes, int n_in,
                              void* d_out, int out_size, void* d_ws, size_t ws_size,
                              hipStream_t stream)
{
  (void)in_sizes; (void)n_in; (void)out_size; (void)ws_size;
  const float* input_seq  = (const float*)d_in[0];   // T x 16
  const float* state_init = (const float*)d_in[1];   // 64
  const float* W_u2x      = (const float*)d_in[2];   // 64 x 80
  const float* b_u2x      = (const float*)d_in[3];   // 64
  const float* W_x2y      = (const float*)d_in[4];   // 1 x 64
  const float* b_x2y      = (const float*)d_in[5];   // 1
  float* out = (float*)d_out;                        // T = 16384 x 16

  float* ws = (float*)d_ws;
  size_t off = 0;
  auto take = [&](size_t n) { float* p = ws + off; off += (n + 63) & ~(size_t)63; return p; };
  float* uProj     = take((size_t)T_STEPS * 64);   // T x 64  == 16384 x 1024
  float* Cc        = take((size_t)NCHUNK * 64);    // chunk carry inputs
  float* Yloc      = take((size_t)NCHUNK * 16);    // local output contributions
  float* SlocFlat  = take((size_t)NSUPER * 1024);  // within-superchunk states
  float* StateFlat = take((size_t)NSUPER * 1024);  // all chunk-boundary states
  float* SPrev     = take((size_t)NCHUNK * 64);    // state entering each chunk
  float* Sbound    = take((size_t)NSUPER * 64);    // state entering each superchunk
  float* Amat   = take(64 * 64);
  float* WuT    = take(16 * 64);
  float* Apow   = take((size_t)17 * 4096);
  float* A16pow = take((size_t)17 * 4096);
  float* qv     = take(17 * 64);
  float* Gt     = take((size_t)1024 * 64);
  float* Lt     = take((size_t)1024 * 16);
  float* Mt     = take((size_t)1024 * 1024);
  float* Rt     = take((size_t)64 * 1024);
  float* Q1t    = take(64 * 16);
  float* by16   = take(16);

  // 0) operator matrices
  rnnscan_setup<<<1, 256, 0, stream>>>(W_u2x, b_u2x, W_x2y, b_x2y,
      Amat, WuT, Apow, A16pow, qv, Gt, Lt, Mt, Rt, Q1t, by16);

  // 1) u_proj = X @ WuT + b_u2x            (262144 x 64, K=16)
  rnnscan_gemm_wmma<4><<<dim3(T_STEPS / 16, 1), 32, 0, stream>>>(
      input_seq, WuT, nullptr, b_u2x, uProj, T_STEPS, 64, 16);

  // 2) chunk carries  Cc = U @ Gt          (16384 x 64, K=1024)
  rnnscan_gemm_wmma<4><<<dim3(NCHUNK / 16, 1), 32, 0, stream>>>(
      uProj, Gt, nullptr, nullptr, Cc, NCHUNK, 64, 1024);

  // 3) local outputs  Yloc = U @ Lt        (16384 x 16, K=1024)
  rnnscan_gemm_wmma<1><<<dim3(NCHUNK / 16, 1), 32, 0, stream>>>(
      uProj, Lt, nullptr, nullptr, Yloc, NCHUNK, 16, 1024);

  // 4) within-superchunk states  SlocFlat = C @ Mt   (1024 x 1024, K=1024)
  rnnscan_gemm_wmma<4><<<dim3(NSUPER / 16, 16), 32, 0, stream>>>(
      Cc, Mt, nullptr, nullptr, SlocFlat, NSUPER, 1024, 1024);

  // 5) residual sequential scan (1024 dependent 64x64 matvecs)
  rnnscan_scan<<<1, 64, 0, stream>>>(A16pow + (size_t)16 * 4096, SlocFlat,
                                     state_init, Sbound);

  // 6) all chunk states  StateFlat = Sbound @ Rt + SlocFlat  (1024 x 1024, K=64)
  rnnscan_gemm_wmma<4><<<dim3(NSUPER / 16, 16), 32, 0, stream>>>(
      Sbound, Rt, SlocFlat, nullptr, StateFlat, NSUPER, 1024, 64);

  // 7) shift by one chunk
  rnnscan_statesprev<<<(NCHUNK * 64 + 255) / 256, 256, 0, stream>>>(
      StateFlat, state_init, SPrev);

  // 8) outputs  Y = SPrev @ Q1t + Yloc + b_y  (16384 x 16, K=64) -> d_out flat
  rnnscan_gemm_wmma<1><<<dim3(NCHUNK / 16, 1), 32, 0, stream>>>(
      SPrev, Q1t, Yloc, by16, out, NCHUNK, 16, 64);
}